// RegressionHeadLoss_62715112456754
// MI455X (gfx1250) — compile-verified
//
#include <hip/hip_runtime.h>
#include <math.h>

// Problem constants (match reference)
#define B_   64
#define N_   32768
#define C_   2
#define K_   10
#define VL_  1.0e6f
#define EPS_ 1e-8f
#define MASKB_ 4.0e6f   // additive bias for masked rows/cols (> VL, keeps invalid)

typedef __attribute__((ext_vector_type(2))) float v2f;
typedef __attribute__((ext_vector_type(8))) float v8f;

__device__ __forceinline__ float sigm(float x) { return 1.0f / (1.0f + expf(-x)); }
__device__ __forceinline__ float softplusf(float x) {
    return fmaxf(x, 0.0f) + log1pf(expf(-fabsf(x)));
}
__device__ __forceinline__ unsigned long long shflxor_u64(unsigned long long v, int m) {
    unsigned int lo = (unsigned int)v;
    unsigned int hi = (unsigned int)(v >> 32);
    lo = (unsigned int)__shfl_xor((int)lo, m, 32);
    hi = (unsigned int)__shfl_xor((int)hi, m, 32);
    return ((unsigned long long)hi << 32) | lo;
}

__global__ __launch_bounds__(1024) void match_loss_kernel(
    const float* __restrict__ p_start, const float* __restrict__ p_end,
    const float* __restrict__ p_cls,   const float* __restrict__ p_conf,
    const float* __restrict__ gt,      float* __restrict__ partial)
{
    __shared__ unsigned int colMatchedW[N_ / 4];   // 32 KB byte flags
    __shared__ float Afs[16], Afe[16], Aoh0[16], Aoh1[16], rowc[16];
    __shared__ unsigned int aliveMask;
    __shared__ unsigned long long bestKey;
    __shared__ int mRows[K_], mCols[K_];
    __shared__ unsigned int mValid[K_];

    const int b    = blockIdx.x;
    const int tid  = threadIdx.x;
    const int lane = tid & 31;
    const int wave = tid >> 5;
    const bool hiHalf = (lane >= 16);
    const int l16  = lane & 15;
    const int nWaves = 1024 >> 5;                  // 32 waves
    const int tilesPerWave = (N_ / 16) / nWaves;   // 64 tiles of 16 columns

    // ---- Phase A: init ----
    for (int i = tid; i < N_ / 4; i += 1024) colMatchedW[i] = 0u;
    if (tid == 0) aliveMask = 0u;
    __syncthreads();
    if (tid < 16) {
        float fs = 0.f, fe = 0.f, o0 = 0.f, o1 = 0.f, rc = MASKB_;
        if (tid < K_) {
            float a = gt[(b * K_ + tid) * 3 + 0];
            float e = gt[(b * K_ + tid) * 3 + 1];
            float c = gt[(b * K_ + tid) * 3 + 2];
            bool pres = !((a != a) || (e != e) || (c != c));
            a = (a != a) ? 0.f : a;
            e = (e != e) ? 0.f : e;
            c = (c != c) ? 0.f : c;
            int ci = (int)c; ci = ci < 0 ? 0 : (ci > C_ - 1 ? C_ - 1 : ci);
            fs = a; fe = e;
            o0 = (ci == 0) ? 1.f : 0.f;
            o1 = (ci == 1) ? 1.f : 0.f;
            rc = fs * fs + fe * fe + 1.0f;         // |onehot|^2 == 1
            if (pres) atomicOr(&aliveMask, 1u << tid);
        }
        Afs[tid] = fs; Afe[tid] = fe; Aoh0[tid] = o0; Aoh1[tid] = o1; rowc[tid] = rc;
    }
    __syncthreads();

    // A operand (16x4 f32): lanes 0-15 -> kdim 0,1 ; lanes 16-31 -> kdim 2,3
    v2f aOp;
    if (!hiHalf) { aOp[0] = Afs[l16];  aOp[1] = Afe[l16];  }
    else         { aOp[0] = Aoh0[l16]; aOp[1] = Aoh1[l16]; }
    const int mBase = hiHalf ? 8 : 0;              // D layout row base for this lane
    float rcL[8];
    #pragma unroll
    for (int r = 0; r < 8; ++r) rcL[r] = rowc[mBase + r];

    const long long baseOff = (long long)b * N_;

    // ---- Phase B: greedy matching, K iterations ----
    for (int it = 0; it < K_; ++it) {
        if (tid == 0) bestKey = ~0ull;
        __syncthreads();
        const unsigned int alive = aliveMask;

        // fold dead/padded/absent rows into the row constant (branchless masking)
        float rcEff[8];
        #pragma unroll
        for (int r = 0; r < 8; ++r)
            rcEff[r] = ((alive >> (mBase + r)) & 1u) ? rcL[r] : MASKB_;

        unsigned long long key = ~0ull;

        for (int t = 0; t < tilesPerWave; ++t) {
            const int tile = wave * tilesPerWave + t;
            const int n = tile * 16 + l16;

            // B operand (4x16 f32): lanes 0-15 load start/end, lanes 16-31 class logits
            float s0, s1;
            if (!hiHalf) {
                s0 = sigm(p_start[baseOff + n]);
                s1 = sigm(p_end[baseOff + n]);
                __builtin_prefetch(&p_start[baseOff + n + 16], 0, 3);
                __builtin_prefetch(&p_end[baseOff + n + 16], 0, 3);
            } else {
                const float2 lg = ((const float2*)p_cls)[baseOff + n];
                s0 = sigm(lg.x);
                s1 = sigm(lg.y);
                __builtin_prefetch(&((const float2*)p_cls)[baseOff + n + 16], 0, 3);
            }
            v2f bOp; bOp[0] = s0; bOp[1] = s1;

            // colconst_n = s^2 + e^2 + sc0^2 + sc1^2 via half-wave exchange
            float pp = s0 * s0 + s1 * s1;
            float cc = pp + __shfl_xor(pp, 16, 32);

            // fold matched-column flag into the cost additively (no EXEC games)
            float colBias =
                (((const unsigned char*)colMatchedW)[n] != 0) ? MASKB_ : 0.0f;
            float base = cc + colBias;

            // cross terms: D = A(16x4) x B(4x16), f32 accumulate
            v8f cz = {};
            v8f d = __builtin_amdgcn_wmma_f32_16x16x4_f32(
                false, aOp, false, bOp, (short)0, cz, false, false);

            #pragma unroll
            for (int r = 0; r < 8; ++r) {
                float cost = fmaxf(fmaf(-2.0f, d[r], rcEff[r] + base), 0.0f);
                unsigned long long kk =
                    (((unsigned long long)__float_as_uint(cost)) << 32) |
                    (unsigned long long)((mBase + r) * N_ + n);
                key = (kk < key) ? kk : key;
            }
        }

        // wave-level min, then one LDS u64 atomic-min per wave
        #pragma unroll
        for (int off = 16; off > 0; off >>= 1) {
            unsigned long long o = shflxor_u64(key, off);
            key = (o < key) ? o : key;
        }
        if (lane == 0) atomicMin(&bestKey, key);
        __syncthreads();

        if (tid == 0) {
            unsigned long long kk = bestKey;
            float v = __uint_as_float((unsigned int)(kk >> 32));
            int idx = (int)(kk & 0xffffffffull);
            int mr = idx / N_, mc = idx % N_;
            bool val = (kk != ~0ull) && (v < VL_ * 0.5f);
            mValid[it] = val ? 1u : 0u;
            mRows[it] = mr; mCols[it] = mc;
            if (val) {
                aliveMask &= ~(1u << mr);
                ((unsigned char*)colMatchedW)[mc] = 1;
            }
        }
        __syncthreads();
    }

    // ---- Phase C: losses on matched pairs (deterministic, thread 0) ----
    if (tid == 0) {
        float num = 0.f, mcount = 0.f;
        for (int i = 0; i < K_; ++i) {
            if (!mValid[i]) continue;
            int k = mRows[i], n = mCols[i];
            float ss = sigm(p_start[baseOff + n]);
            float se = sigm(p_end[baseOff + n]);
            float gs = Afs[k], ge = Afe[k];
            float loc = (ss - gs) * (ss - gs) + (se - ge) * (se - ge);

            float l0 = p_cls[(baseOff + n) * C_ + 0];
            float l1 = p_cls[(baseOff + n) * C_ + 1];
            float y0 = Aoh0[k], y1 = Aoh1[k];
            float bce = y0 * softplusf(-l0) + (1.f - y0) * softplusf(l0)
                      + y1 * softplusf(-l1) + (1.f - y1) * softplusf(l1);

            float a1 = fminf(ss, se), b1v = fmaxf(ss, se);
            float a2 = fminf(gs, ge), b2v = fmaxf(gs, ge);
            float inter = fmaxf(0.f, fminf(b1v, b2v) - fmaxf(a1, a2));
            float uni   = fmaxf(EPS_, fmaxf(b1v, b2v) - fminf(a1, a2));
            float iou = inter / uni;
            float pc = sigm(p_conf[baseOff + n]);
            float conf = (pc - iou) * (pc - iou);

            num += loc + bce + conf;
            mcount += 1.f;
        }
        partial[2 * b + 0] = num;
        partial[2 * b + 1] = mcount;
    }
}

__global__ void finalize_kernel(const float* __restrict__ partial, float* __restrict__ out) {
    if (threadIdx.x == 0 && blockIdx.x == 0) {
        float num = 0.f, m = 0.f;
        for (int i = 0; i < B_; ++i) { num += partial[2 * i]; m += partial[2 * i + 1]; }
        out[0] = (m > 0.f) ? num / (m + EPS_) : 0.f;
    }
}

extern "C" void kernel_launch(void* const* d_in, const int* in_sizes, int n_in,
                              void* d_out, int out_size, void* d_ws, size_t ws_size,
                              hipStream_t stream) {
    const float* p_start = (const float*)d_in[0];
    const float* p_end   = (const float*)d_in[1];
    const float* p_cls   = (const float*)d_in[2];
    const float* p_conf  = (const float*)d_in[3];
    const float* gt      = (const float*)d_in[4];
    float* partial = (float*)d_ws;   // 64 * 2 floats

    match_loss_kernel<<<B_, 1024, 0, stream>>>(p_start, p_end, p_cls, p_conf, gt, partial);
    finalize_kernel<<<1, 32, 0, stream>>>(partial, (float*)d_out);
}